// SpanTreeSplitGNN_32049045962864
// MI455X (gfx1250) — compile-verified
//
#include <hip/hip_runtime.h>

typedef float v2f __attribute__((ext_vector_type(2)));
typedef float v8f __attribute__((ext_vector_type(8)));
typedef uint32_t u32x4 __attribute__((ext_vector_type(4)));
typedef uint32_t u32x8 __attribute__((ext_vector_type(8)));

#define N_NODES 20000
#define HDIM    128
#define GGROUPS 64
#define EPS_GN  1e-5f
#define SLOPE   0.01f

// ------------------------------------------------------------------
// C[M x 128] = A[M x 128] @ W[128 x 128] (+ optional bias per column)
// fp32 WMMA 16x16x4. One wave -> 16 rows x 128 cols (8 v8f accums).
// W staged into LDS via TDM (rows 0..63) + async global->LDS (rows 64..127).
// ------------------------------------------------------------------
__global__ void gemm128_wmma(const float* __restrict__ A,
                             const float* __restrict__ W,
                             const float* __restrict__ bias,
                             float* __restrict__ C, int M) {
  __shared__ float Wlds[128 * 128];

  {
    const uint32_t ldsBase = (uint32_t)(uintptr_t)(void*)&Wlds[0];
    const uint64_t gbase = (uint64_t)(uintptr_t)W;

    // --- wave 0: Tensor Data Mover copies rows 0..63 (32 KB) ---------------
    if (threadIdx.x < 32) {
      u32x4 g0;
      g0.x = 1u;                                    // count=1 (valid descriptor)
      g0.y = ldsBase;                               // lds_addr (bytes)
      g0.z = (uint32_t)gbase;                       // global_addr[31:0]
      g0.w = ((uint32_t)(gbase >> 32) & 0x01FFFFFFu) | (2u << 30); // addr[56:32] | type=2
      u32x8 g1;
      g1[0] = 0x00020000u;   // workgroup_mask=0, data_size=2 (4 bytes)
      g1[1] = 128u << 16;    // tensor_dim0[15:0]=128 in bits 63:48
      g1[2] = 64u << 16;     // tensor_dim0[31:16]=0 | tensor_dim1[15:0]=64
      g1[3] = 128u << 16;    // tensor_dim1[31:16]=0 | tile_dim0=128
      g1[4] = 64u;           // tile_dim1=64, tile_dim2=0
      g1[5] = 128u;          // tensor_dim0_stride[31:0]=128
      g1[6] = 0u;            // stride hi / tensor_dim1_stride lo
      g1[7] = 0u;            // tensor_dim1_stride hi
      asm volatile("tensor_load_to_lds %0, %1" :: "s"(g0), "s"(g1) : "memory");
    }

    // --- all waves: async DMA rows 64..127 (32 KB): 8 x b128 per thread ----
#pragma unroll
    for (int it = 0; it < 8; ++it) {
      const uint32_t off = 32768u + (uint32_t)(it * 256 + threadIdx.x) * 16u; // bytes
      const uint32_t ldsAddr = ldsBase + off;
      const uint64_t gaddr = gbase + (uint64_t)off;
      asm volatile("global_load_async_to_lds_b128 %0, %1, off"
                   :: "v"(ldsAddr), "v"(gaddr)
                   : "memory");
    }
    asm volatile("s_wait_asynccnt 0" ::: "memory");
    if (threadIdx.x < 32) __builtin_amdgcn_s_wait_tensorcnt(0);
  }
  __syncthreads();   // all waves arrived => all TDM + async copies complete

  const int wave = threadIdx.x >> 5;
  const int lane = threadIdx.x & 31;
  const int half = lane >> 4;     // 0: lanes 0-15, 1: lanes 16-31
  const int lm   = lane & 15;
  const int tileRow = blockIdx.x * 8 + wave;
  const int r0 = tileRow * 16;
  if (r0 >= M) return;            // wave-uniform exit

  v8f acc[8] = {};

  const float* __restrict__ arow = A + (size_t)(r0 + lm) * HDIM;
  for (int k0 = 0; k0 < 128; k0 += 4) {
    // A 16x4 frag: lanes 0-15 -> K=k0,k0+1 ; lanes 16-31 -> K=k0+2,k0+3
    v2f a;
    a.x = arow[k0 + 2 * half + 0];
    a.y = arow[k0 + 2 * half + 1];
#pragma unroll
    for (int t = 0; t < 8; ++t) {
      const int c0 = t * 16;
      // B 4x16 frag mirrored across lane halves
      v2f b;
      b.x = Wlds[(k0 + 2 * half + 0) * 128 + c0 + lm];
      b.y = Wlds[(k0 + 2 * half + 1) * 128 + c0 + lm];
      acc[t] = __builtin_amdgcn_wmma_f32_16x16x4_f32(
          false, a, false, b, (short)0, acc[t], false, false);
    }
  }

#pragma unroll
  for (int t = 0; t < 8; ++t) {
    const int c0 = t * 16;
    const float bb = bias ? bias[c0 + lm] : 0.0f;
#pragma unroll
    for (int i = 0; i < 8; ++i) {
      const int row = r0 + i + 8 * half;   // VGPR i: M=i (lo half) / M=i+8 (hi half)
      C[(size_t)row * HDIM + c0 + lm] = acc[t][i] + bb;
    }
  }
}

// ------------------------------------------------------------------
// Per-group node counts, then cnt=max(cnt,1) and rc = 1/(cnt*L)
// ------------------------------------------------------------------
__global__ void count_nodes(const int* __restrict__ batch, float* cntraw, int n) {
  int i = blockIdx.x * blockDim.x + threadIdx.x;
  if (i < n) atomicAdd(&cntraw[batch[i]], 1.0f);
}
__global__ void finalize_cnt(float* cntraw, float* rc) {
  int g = threadIdx.x;   // 64 threads
  float c = fmaxf(cntraw[g], 1.0f);
  cntraw[g] = c;
  rc[g] = 1.0f / (c * 3.0f);   // includes /L for layer averaging
}

// ------------------------------------------------------------------
// Degree + dinv (in place: buf starts zeroed, accum counts, then rsqrt)
// ------------------------------------------------------------------
__global__ void deg_accum(const int* __restrict__ dst, float* deg, int e) {
  int i = blockIdx.x * blockDim.x + threadIdx.x;
  if (i < e) atomicAdd(&deg[dst[i]], 1.0f);
}
__global__ void deg_to_dinv(float* deg, int n) {
  int i = blockIdx.x * blockDim.x + threadIdx.x;
  if (i < n) deg[i] = rsqrtf(deg[i] + 1.0f);
}

// ------------------------------------------------------------------
// out[i] = h[i]*dinv[i]^2 + b   (GCN self loop term + bias)
// one wave per node, 4 channels per lane
// ------------------------------------------------------------------
__global__ void self_init(const float* __restrict__ h, const float* __restrict__ dinv,
                          const float* __restrict__ b, float* __restrict__ out, int n) {
  int t = blockIdx.x * blockDim.x + threadIdx.x;
  int node = t >> 5;
  int c4 = (t & 31) * 4;
  if (node >= n) return;
  float di = dinv[node];
  float s = di * di;
  const float4 hv = *(const float4*)(h + (size_t)node * HDIM + c4);
  float4 o;
  o.x = hv.x * s + b[c4 + 0];
  o.y = hv.y * s + b[c4 + 1];
  o.z = hv.z * s + b[c4 + 2];
  o.w = hv.w * s + b[c4 + 3];
  *(float4*)(out + (size_t)node * HDIM + c4) = o;
}

// ------------------------------------------------------------------
// out[dst] += h[src] * dinv[src]*dinv[dst] : one wave per edge
// ------------------------------------------------------------------
__global__ void propagate(const float* __restrict__ h, const float* __restrict__ dinv,
                          const int* __restrict__ src, const int* __restrict__ dst,
                          float* __restrict__ out, int e) {
  int wid = (blockIdx.x * blockDim.x + threadIdx.x) >> 5;
  int lane = threadIdx.x & 31;
  if (wid >= e) return;
  int s = src[wid];
  int d = dst[wid];
  float coef = dinv[s] * dinv[d];
  const float4 hv = *(const float4*)(h + (size_t)s * HDIM + lane * 4);
  float* op = out + (size_t)d * HDIM + lane * 4;
  atomicAdd(op + 0, hv.x * coef);
  atomicAdd(op + 1, hv.y * coef);
  atomicAdd(op + 2, hv.z * coef);
  atomicAdd(op + 3, hv.w * coef);
}

// ------------------------------------------------------------------
// GraphNorm stats: per (group, channel) sum and sum-of-squares
// ------------------------------------------------------------------
__global__ void grp_stats(const float* __restrict__ x, const int* __restrict__ batch,
                          float* __restrict__ sums, float* __restrict__ sumsq, int n) {
  int t = blockIdx.x * blockDim.x + threadIdx.x;
  int node = t >> 5;
  int c4 = (t & 31) * 4;
  if (node >= n) return;
  int g = batch[node];
  const float4 v = *(const float4*)(x + (size_t)node * HDIM + c4);
  float* sp = sums + g * HDIM + c4;
  float* qp = sumsq + g * HDIM + c4;
  atomicAdd(sp + 0, v.x); atomicAdd(qp + 0, v.x * v.x);
  atomicAdd(sp + 1, v.y); atomicAdd(qp + 1, v.y * v.y);
  atomicAdd(sp + 2, v.z); atomicAdd(qp + 2, v.z * v.z);
  atomicAdd(sp + 3, v.w); atomicAdd(qp + 3, v.w * v.w);
}

// var = E[x^2] - mean^2 * ms * (2 - ms)   (single-pass GraphNorm)
__global__ void finalize_stats(const float* __restrict__ sums, const float* __restrict__ sumsq,
                               const float* __restrict__ cnt, const float* __restrict__ gms,
                               float* __restrict__ meanms, float* __restrict__ invv) {
  int t = blockIdx.x * blockDim.x + threadIdx.x;   // 8192
  int g = t >> 7;
  int c = t & 127;
  float ic = 1.0f / cnt[g];
  float mean = sums[t] * ic;
  float ms = gms[c];
  float var = sumsq[t] * ic - mean * mean * ms * (2.0f - ms);
  meanms[t] = ms * mean;
  invv[t] = rsqrtf(var + EPS_GN);
}

// ------------------------------------------------------------------
// y = lrelu(gw*(x - ms*mean)*inv + gb); write cur; accumulate group mean
// into acc[64 x 640] at column base (folds /cnt and /L via rc)
// ------------------------------------------------------------------
__global__ void normalize_lrelu_acc(const float* __restrict__ x, const int* __restrict__ batch,
                                    const float* __restrict__ meanms, const float* __restrict__ invv,
                                    const float* __restrict__ gw, const float* __restrict__ gb,
                                    const float* __restrict__ rc, float* __restrict__ outcur,
                                    float* __restrict__ acc, int colbase, int n) {
  int t = blockIdx.x * blockDim.x + threadIdx.x;
  int node = t >> 5;
  int c4 = (t & 31) * 4;
  if (node >= n) return;
  int g = batch[node];
  float scale = rc[g];
  const float4 v = *(const float4*)(x + (size_t)node * HDIM + c4);
  float vv[4] = {v.x, v.y, v.z, v.w};
  float y[4];
#pragma unroll
  for (int j = 0; j < 4; ++j) {
    int c = c4 + j;
    float o = (vv[j] - meanms[g * HDIM + c]) * invv[g * HDIM + c];
    o = gw[c] * o + gb[c];
    y[j] = o > 0.0f ? o : SLOPE * o;
  }
  float4 yo = {y[0], y[1], y[2], y[3]};
  *(float4*)(outcur + (size_t)node * HDIM + c4) = yo;
  float* ap = acc + g * 640 + colbase + c4;
  atomicAdd(ap + 0, y[0] * scale);
  atomicAdd(ap + 1, y[1] * scale);
  atomicAdd(ap + 2, y[2] * scale);
  atomicAdd(ap + 3, y[3] * scale);
}

// ------------------------------------------------------------------
// out[64 x 128] = acc[64 x 640] @ merge_W[640 x 128] + merge_b
// ------------------------------------------------------------------
__global__ void merge_out(const float* __restrict__ acc, const float* __restrict__ mW,
                          const float* __restrict__ mb, float* __restrict__ out) {
  int t = blockIdx.x * blockDim.x + threadIdx.x;   // 8192
  int g = t >> 7;
  int j = t & 127;
  float s = mb[j];
  const float* arow = acc + g * 640;
  for (int k = 0; k < 640; ++k) s += arow[k] * mW[k * 128 + j];
  out[t] = s;
}

// ------------------------------------------------------------------
extern "C" void kernel_launch(void* const* d_in, const int* in_sizes, int n_in,
                              void* d_out, int out_size, void* d_ws, size_t ws_size,
                              hipStream_t stream) {
  const float* x        = (const float*)d_in[0];
  const float* emb_W    = (const float*)d_in[1];
  const float* emb_b    = (const float*)d_in[2];
  const float* main_W   = (const float*)d_in[3];
  const float* main_b   = (const float*)d_in[4];
  const float* main_gw  = (const float*)d_in[5];
  const float* main_gb  = (const float*)d_in[6];
  const float* main_gms = (const float*)d_in[7];
  const float* loc_W    = (const float*)d_in[8];
  const float* loc_b    = (const float*)d_in[9];
  const float* loc_gw   = (const float*)d_in[10];
  const float* loc_gb   = (const float*)d_in[11];
  const float* loc_gms  = (const float*)d_in[12];
  const float* merge_W  = (const float*)d_in[13];
  const float* merge_b  = (const float*)d_in[14];
  const int* edge_index = (const int*)d_in[15];
  const int* e0         = (const int*)d_in[16];
  const int* e1         = (const int*)d_in[17];
  const int* e2         = (const int*)d_in[18];
  const int* e3         = (const int*)d_in[19];
  const int* batch      = (const int*)d_in[20];

  const int N = N_NODES;

  // workspace carve (floats)
  float* w = (float*)d_ws;
  float* h_emb  = w; w += (size_t)N * HDIM;
  float* cur    = w; w += (size_t)N * HDIM;
  float* hbuf   = w; w += (size_t)N * HDIM;
  float* outbuf = w; w += (size_t)N * HDIM;
  float* dinv   = w; w += N;
  float* cntraw = w; w += GGROUPS;
  float* rc     = w; w += GGROUPS;
  float* sums   = w; w += GGROUPS * HDIM;
  float* sumsq  = w; w += GGROUPS * HDIM;
  float* meanms = w; w += GGROUPS * HDIM;
  float* invv   = w; w += GGROUPS * HDIM;
  float* acc    = w; w += GGROUPS * 640;

  auto cdiv = [](int a, int b) { return (a + b - 1) / b; };

  // --- group counts + merge accumulator init ---
  hipMemsetAsync(cntraw, 0, 2 * GGROUPS * sizeof(float), stream);
  hipMemsetAsync(acc, 0, (size_t)GGROUPS * 640 * sizeof(float), stream);
  count_nodes<<<cdiv(N, 256), 256, 0, stream>>>(batch, cntraw, N);
  finalize_cnt<<<1, GGROUPS, 0, stream>>>(cntraw, rc);

  // --- embedding: h = x @ emb_W + emb_b ---
  const int gemmBlocks = cdiv(N / 16, 8);
  gemm128_wmma<<<gemmBlocks, 256, 0, stream>>>(x, emb_W, emb_b, h_emb, N);

  // --- branches in concat order: edges0..3 (loc weights), then main ---
  struct Branch {
    const int* ei; int ecnt;
    const float* W; const float* b;
    const float* gw; const float* gb; const float* gms;
    int colbase;
  };
  Branch br[5] = {
    {e0,         in_sizes[16] / 2, loc_W,  loc_b,  loc_gw,  loc_gb,  loc_gms,  0},
    {e1,         in_sizes[17] / 2, loc_W,  loc_b,  loc_gw,  loc_gb,  loc_gms,  128},
    {e2,         in_sizes[18] / 2, loc_W,  loc_b,  loc_gw,  loc_gb,  loc_gms,  256},
    {e3,         in_sizes[19] / 2, loc_W,  loc_b,  loc_gw,  loc_gb,  loc_gms,  384},
    {edge_index, in_sizes[15] / 2, main_W, main_b, main_gw, main_gb, main_gms, 512},
  };

  for (int bi = 0; bi < 5; ++bi) {
    const Branch& B = br[bi];
    const int E = B.ecnt;
    const int* src = B.ei;
    const int* dst = B.ei + E;

    // dinv = rsqrt(deg+1)
    hipMemsetAsync(dinv, 0, (size_t)N * sizeof(float), stream);
    deg_accum<<<cdiv(E, 256), 256, 0, stream>>>(dst, dinv, E);
    deg_to_dinv<<<cdiv(N, 256), 256, 0, stream>>>(dinv, N);

    for (int l = 0; l < 3; ++l) {
      const float* xin = (l == 0) ? h_emb : cur;
      // h = x @ W_l
      gemm128_wmma<<<gemmBlocks, 256, 0, stream>>>(xin, B.W + (size_t)l * HDIM * HDIM,
                                                   nullptr, hbuf, N);
      // out = h*dinv^2 + b ; out[dst] += h[src]*dinv[src]*dinv[dst]
      self_init<<<cdiv(N * 32, 256), 256, 0, stream>>>(hbuf, dinv, B.b + l * HDIM, outbuf, N);
      propagate<<<cdiv(E * 32, 256), 256, 0, stream>>>(hbuf, dinv, src, dst, outbuf, E);
      // GraphNorm (single stats pass) + leaky relu + group-mean accumulation
      hipMemsetAsync(sums, 0, 2 * GGROUPS * HDIM * sizeof(float), stream);
      grp_stats<<<cdiv(N * 32, 256), 256, 0, stream>>>(outbuf, batch, sums, sumsq, N);
      finalize_stats<<<cdiv(GGROUPS * HDIM, 256), 256, 0, stream>>>(sums, sumsq, cntraw,
                                                                    B.gms + l * HDIM, meanms, invv);
      normalize_lrelu_acc<<<cdiv(N * 32, 256), 256, 0, stream>>>(outbuf, batch, meanms, invv,
                                                                 B.gw + l * HDIM, B.gb + l * HDIM,
                                                                 rc, cur, acc, B.colbase, N);
    }
  }

  // --- final tiny merge GEMM: out = acc @ merge_W + merge_b ---
  merge_out<<<cdiv(GGROUPS * HDIM, 128), 128, 0, stream>>>(acc, merge_W, merge_b, (float*)d_out);
}